// GEBLNet_86414741996362
// MI455X (gfx1250) — compile-verified
//
#include <hip/hip_runtime.h>
#include <hip/hip_bf16.h>
#include <math.h>

// GEBLNet fused kernel for gfx1250 (MI455X).
//  - u-channels never influence the output -> dropped.
//  - layer2 only needs traces: tr(y2_u) = sum_{vw} w2[u,v,w] * T[v,w],
//    T[v,w] = sum_{ik} W_v[i,k] W_w[k,i] (complex). The K=625 contraction
//    tr_u[p] = sum_k T[p,k] * w2T[k,u] runs on V_WMMA_F32_16X16X4_F32:
//    M = 16 points, N = 12 u (pad 16), K = one v-row at a time (25 -> 28
//    = 7 K-steps), so no runtime divisions in any hot loop.
//  - conj channels are never materialized: the T loop needs W_w[k,i]; for
//    conj channels that is conj(W_{w-12}[i,k]) -> plain read, negated imag.
//    Identity column = trace(A).
//  - w1 staged per block into LDS as [v][w][u][2] (contiguous DS reads).
//  - layer-1 register pressure bounded: u-tile of 4 (3 passes) + P fused
//    into the contraction -> ~130 live floats, no scratch spills; unrolling
//    explicitly capped.
//  - 64-thread blocks (2 waves), ~99 KB LDS/block -> 3 blocks/WGP.

typedef float v2f __attribute__((ext_vector_type(2)));
typedef float v8f __attribute__((ext_vector_type(8)));

#define NPB 64             // threads per block = points per block, 2 waves
#define W1F 4056           // floats: staged w1, [v][w][u][2] = 13*13*12*2
#define H1F 6912           // floats: 12ch * 9 * 32pts * 2
#define SCRF 3456          // floats: x-stage (3456) / T-rows (1848) / C (768)
#define PW  (H1F + SCRF)   // per-wave floats

// ---- branch-free matrix loaders (SoA: buf[((ch*9+e)*32+lane)*2 + c]) ------
__device__ __forceinline__ void load_mat(const float* buf, int ch, int lane,
                                         float R[9], float I[9]) {
#pragma unroll
  for (int e = 0; e < 9; ++e) {
    int b = ((ch * 9 + e) * 32 + lane) * 2;
    R[e] = buf[b]; I[e] = buf[b + 1];
  }
}
__device__ __forceinline__ void load_mat_conj(const float* buf, int ch, int lane,
                                              float R[9], float I[9]) {
#pragma unroll
  for (int i = 0; i < 3; ++i)
#pragma unroll
    for (int k = 0; k < 3; ++k) {
      int b = ((ch * 9 + k * 3 + i) * 32 + lane) * 2;
      R[i * 3 + k] = buf[b]; I[i * 3 + k] = -buf[b + 1];
    }
}
__device__ __forceinline__ void load_mat_trans(const float* buf, int ch, int lane,
                                               float R[9], float I[9]) {
#pragma unroll
  for (int i = 0; i < 3; ++i)
#pragma unroll
    for (int k = 0; k < 3; ++k) {
      int b = ((ch * 9 + k * 3 + i) * 32 + lane) * 2;
      R[i * 3 + k] = buf[b]; I[i * 3 + k] = buf[b + 1];
    }
}
__device__ __forceinline__ void load_mat_negim(const float* buf, int ch, int lane,
                                               float R[9], float I[9]) {
#pragma unroll
  for (int e = 0; e < 9; ++e) {
    int b = ((ch * 9 + e) * 32 + lane) * 2;
    R[e] = buf[b]; I[e] = -buf[b + 1];
  }
}
__device__ __forceinline__ void load_id(float R[9], float I[9]) {
#pragma unroll
  for (int e = 0; e < 9; ++e) {
    R[e] = (e == 0 || e == 4 || e == 8) ? 1.f : 0.f; I[e] = 0.f;
  }
}
// acc[uu] += w1[pass*4+uu, v, w] * P with given P (identity fast paths)
__device__ __forceinline__ void contract4(const float* W1S, int pass, int v,
                                          int w, const float Pr[9],
                                          const float Pi[9],
                                          float acc[4][9][2]) {
  const float* wrow = W1S + ((v * 13 + w) * 12 + pass * 4) * 2;
#pragma unroll
  for (int uu = 0; uu < 4; ++uu) {
    float wr = wrow[uu * 2], wi = wrow[uu * 2 + 1];
#pragma unroll
    for (int e = 0; e < 9; ++e) {
      acc[uu][e][0] += wr * Pr[e] - wi * Pi[e];
      acc[uu][e][1] += wr * Pi[e] + wi * Pr[e];
    }
  }
}
// fused P = A@B (complex 3x3) + 4-channel contraction; no P arrays live
__device__ __forceinline__ void fused4(const float* W1S, int pass, int v, int w,
                                       const float Ar[9], const float Ai[9],
                                       const float Br[9], const float Bi[9],
                                       float acc[4][9][2]) {
  const float* wrow = W1S + ((v * 13 + w) * 12 + pass * 4) * 2;
  float wr[4], wi[4];
#pragma unroll
  for (int uu = 0; uu < 4; ++uu) { wr[uu] = wrow[uu * 2]; wi[uu] = wrow[uu * 2 + 1]; }
#pragma unroll
  for (int i = 0; i < 3; ++i)
#pragma unroll
    for (int j = 0; j < 3; ++j) {
      float pr = 0.f, pi = 0.f;
#pragma unroll
      for (int k = 0; k < 3; ++k) {
        float ar = Ar[i * 3 + k], ai = Ai[i * 3 + k];
        float br = Br[k * 3 + j], bi = Bi[k * 3 + j];
        pr += ar * br - ai * bi;
        pi += ar * bi + ai * br;
      }
      int e = i * 3 + j;
#pragma unroll
      for (int uu = 0; uu < 4; ++uu) {
        acc[uu][e][0] += wr[uu] * pr - wi[uu] * pi;
        acc[uu][e][1] += wr[uu] * pi + wi[uu] * pr;
      }
    }
}
// T[v,w] = sum_e A[e] * B[e] (complex) -> SCR slot
__device__ __forceinline__ void tdot_store(float* SCR, int slot, int lane,
                                           const float Ar[9], const float Ai[9],
                                           const float Br[9], const float Bi[9]) {
  float tre = 0.f, tim = 0.f;
#pragma unroll
  for (int e = 0; e < 9; ++e) {
    tre += Ar[e] * Br[e] - Ai[e] * Bi[e];
    tim += Ar[e] * Bi[e] + Ai[e] * Br[e];
  }
  SCR[slot * 66 + lane]      = tre;
  SCR[slot * 66 + 33 + lane] = tim;
}

__global__ void geblnet_fused(const float* __restrict__ xg,
                              const float* __restrict__ w1g,
                              const float* __restrict__ w2g,
                              const float* __restrict__ dwg,
                              const float* __restrict__ dbg,
                              float* __restrict__ outg,
                              int npts) {
  extern __shared__ float smem[];
  float* W1S = smem;                       // block-shared staged w1
  const int wv   = threadIdx.x >> 5;
  const int lane = threadIdx.x & 31;
  float* H1  = smem + W1F + wv * PW;       // 12 base channels (layer-2 input)
  float* SCR = H1 + H1F;                   // x-stage / T rows / C dump

  const int pt0 = (blockIdx.x * (blockDim.x >> 5) + wv) * 32;
  const int pt  = pt0 + lane;
  __builtin_prefetch(xg + (size_t)pt * 180 + 72, 0, 1);

  // -------- Stage w1 -> LDS transposed [v][w][u][2] ------------------------
#pragma unroll 1
  for (int idx = threadIdx.x; idx < 2028; idx += NPB) {   // 12*13*13 pairs
    int u = idx / 169;
    int r = idx - u * 169;
    int v = r / 13;
    int w = r - v * 13;
    int dst = ((v * 13 + w) * 12 + u) * 2;
    W1S[dst]     = w1g[idx * 2];
    W1S[dst + 1] = w1g[idx * 2 + 1];
  }

  // -------- Stage 0: cooperative load of the 6 input w-channels -> SCR -----
  {
    int p = 0, e = lane;                      // element walker, single carry
#pragma unroll 1
    for (int it = 0; it < 108; ++it) {        // 108*32 = 3456 elements
      int ee = e;
      int s0 = (ee >= 54) ? 3 : 0; ee -= s0 * 18;
      int s1 = (ee >= 36) ? 2 : ((ee >= 18) ? 1 : 0); ee -= s1 * 18;
      int s = s0 + s1;                        // channel = e/18, branch-free
      SCR[((s * 9 + (ee >> 1)) * 32 + p) * 2 + (ee & 1)] =
          xg[(size_t)(pt0 + p) * 180 + 72 + e];
      e += 32;
      if (e >= 108) { e -= 108; p += 1; }
    }
  }
  __syncthreads();                            // W1S visible to both waves

  // -------- Layer 1: segmented gebl(w1) + gerelu, u tiled 3 x 4 ------------
  float sumabs = 0.f;
#pragma unroll 1
  for (int pass = 0; pass < 3; ++pass) {
    float acc[4][9][2];
#pragma unroll
    for (int a = 0; a < 4; ++a)
#pragma unroll
      for (int e = 0; e < 9; ++e) { acc[a][e][0] = 0.f; acc[a][e][1] = 0.f; }

    float Ar[9], Ai[9], Br[9], Bi[9];
#pragma unroll 1
    for (int vs = 0; vs < 6; ++vs) {          // A base segment
      load_mat(SCR, vs, lane, Ar, Ai);
#pragma unroll 1
      for (int ws = 0; ws < 6; ++ws) {
        load_mat(SCR, ws, lane, Br, Bi);
        fused4(W1S, pass, vs, ws, Ar, Ai, Br, Bi, acc);
      }
#pragma unroll 1
      for (int ws = 0; ws < 6; ++ws) {
        load_mat_conj(SCR, ws, lane, Br, Bi);
        fused4(W1S, pass, vs, ws + 6, Ar, Ai, Br, Bi, acc);
      }
      contract4(W1S, pass, vs, 12, Ar, Ai, acc);       // B = I -> P = A
    }
#pragma unroll 1
    for (int vs = 0; vs < 6; ++vs) {          // A conj segment
      load_mat_conj(SCR, vs, lane, Ar, Ai);
#pragma unroll 1
      for (int ws = 0; ws < 6; ++ws) {
        load_mat(SCR, ws, lane, Br, Bi);
        fused4(W1S, pass, vs + 6, ws, Ar, Ai, Br, Bi, acc);
      }
#pragma unroll 1
      for (int ws = 0; ws < 6; ++ws) {
        load_mat_conj(SCR, ws, lane, Br, Bi);
        fused4(W1S, pass, vs + 6, ws + 6, Ar, Ai, Br, Bi, acc);
      }
      contract4(W1S, pass, vs + 6, 12, Ar, Ai, acc);   // B = I -> P = A
    }
#pragma unroll 1
    for (int ws = 0; ws < 6; ++ws) {          // A = I row: P = B
      load_mat(SCR, ws, lane, Br, Bi);
      contract4(W1S, pass, 12, ws, Br, Bi, acc);
    }
#pragma unroll 1
    for (int ws = 0; ws < 6; ++ws) {
      load_mat_conj(SCR, ws, lane, Br, Bi);
      contract4(W1S, pass, 12, ws + 6, Br, Bi, acc);
    }
    load_id(Br, Bi);
    contract4(W1S, pass, 12, 12, Br, Bi, acc);         // I @ I = I

    // gerelu scale per channel; store to H1; accumulate |trace|
#pragma unroll
    for (int uu = 0; uu < 4; ++uu) {
      int u = pass * 4 + uu;
      float trr = acc[uu][0][0] + acc[uu][4][0] + acc[uu][8][0];
      float tri = acc[uu][0][1] + acc[uu][4][1] + acc[uu][8][1];
      float s = fmaxf(trr, 0.f);
#pragma unroll
      for (int e = 0; e < 9; ++e) {
        int b = ((u * 9 + e) * 32 + lane) * 2;
        H1[b]     = s * acc[uu][e][0];
        H1[b + 1] = s * acc[uu][e][1];
      }
      sumabs += s * sqrtf(trr * trr + tri * tri);
    }
  }
  // trnorm: rescale H1 in place
  float inv1 = 1.f / fmaxf(sumabs * (1.f / 12.f), 0.001f);
#pragma unroll 1
  for (int e2 = 0; e2 < 108; ++e2) {
    int b = (e2 * 32 + lane) * 2;
    H1[b] *= inv1; H1[b + 1] *= inv1;
  }

  // -------- Layer 2: T rows + WMMA contraction (K = 25 rows x 28) ----------
  v8f cre[2], cim[2];
#pragma unroll
  for (int t = 0; t < 2; ++t) {
    cre[t] = (v8f){0.f,0.f,0.f,0.f,0.f,0.f,0.f,0.f};
    cim[t] = (v8f){0.f,0.f,0.f,0.f,0.f,0.f,0.f,0.f};
  }
  const int m = lane & 15, h = lane >> 4;
#pragma unroll
  for (int w = 25; w < 28; ++w) {                      // zero K padding
    SCR[w * 66 + lane] = 0.f; SCR[w * 66 + 33 + lane] = 0.f;
  }

#pragma unroll 1
  for (int v = 0; v < 25; ++v) {
    float Ar[9], Ai[9];
    if (v < 12)      load_mat(H1, v, lane, Ar, Ai);    // uniform 3-way
    else if (v < 24) load_mat_conj(H1, v - 12, lane, Ar, Ai);
    else             load_id(Ar, Ai);

    // slots 0..11: need W_w[k,i] -> transposed read of base channel
#pragma unroll 2
    for (int w = 0; w < 12; ++w) {
      float Br[9], Bi[9];
      load_mat_trans(H1, w, lane, Br, Bi);
      tdot_store(SCR, w, lane, Ar, Ai, Br, Bi);
    }
    // slots 12..23: W_w = conj(base^T) -> W_w[k,i] = conj(base[i,k])
#pragma unroll 2
    for (int w = 0; w < 12; ++w) {
      float Br[9], Bi[9];
      load_mat_negim(H1, w, lane, Br, Bi);
      tdot_store(SCR, w + 12, lane, Ar, Ai, Br, Bi);
    }
    // slot 24: identity column -> T = trace(A)
    SCR[24 * 66 + lane]      = Ar[0] + Ar[4] + Ar[8];
    SCR[24 * 66 + 33 + lane] = Ai[0] + Ai[4] + Ai[8];

    // 7 K-steps of 4 over this row; B[k][n] = w2[n, v, k] (zero-padded)
#pragma unroll
    for (int kq = 0; kq < 7; ++kq) {
      int k0 = kq * 4 + 2 * h;                         // lane-half K rows
      int k1 = k0 + 1;
      int nc  = m < 12 ? m : 11;
      int w0c = k0 < 25 ? k0 : 24;
      int w1c = k1 < 25 ? k1 : 24;
      float ok0 = (k0 < 25 && m < 12) ? 1.f : 0.f;
      float ok1 = (k1 < 25 && m < 12) ? 1.f : 0.f;
      int i0 = ((nc * 25 + v) * 25 + w0c) * 2;
      int i1 = ((nc * 25 + v) * 25 + w1c) * 2;
      float wr0 = w2g[i0] * ok0, wi0 = w2g[i0 + 1] * ok0;
      float wr1 = w2g[i1] * ok1, wi1 = w2g[i1 + 1] * ok1;
      v2f br  = {wr0, wr1};
      v2f bi  = {wi0, wi1};
      v2f bni = {-wi0, -wi1};
#pragma unroll
      for (int t = 0; t < 2; ++t) {
        int p = t * 16 + m;                            // M row = point
        v2f ar = { SCR[k0 * 66 + p],      SCR[k1 * 66 + p] };
        v2f ai = { SCR[k0 * 66 + 33 + p], SCR[k1 * 66 + 33 + p] };
        // complex MMA: re += Tr*Wr - Ti*Wi ; im += Tr*Wi + Ti*Wr
        cre[t] = __builtin_amdgcn_wmma_f32_16x16x4_f32(false, ar, false, br,  (short)0, cre[t], false, false);
        cre[t] = __builtin_amdgcn_wmma_f32_16x16x4_f32(false, ai, false, bni, (short)0, cre[t], false, false);
        cim[t] = __builtin_amdgcn_wmma_f32_16x16x4_f32(false, ar, false, bi,  (short)0, cim[t], false, false);
        cim[t] = __builtin_amdgcn_wmma_f32_16x16x4_f32(false, ai, false, br,  (short)0, cim[t], false, false);
      }
    }
  }

  // -------- Dump C tiles (lane = N col, vgpr = M row) -> per-point layout --
#pragma unroll
  for (int t = 0; t < 2; ++t)
#pragma unroll
    for (int g = 0; g < 8; ++g) {
      int p = t * 16 + g + 8 * h;
      if (m < 12) {
        SCR[p * 24 + m * 2]     = cre[t][g];
        SCR[p * 24 + m * 2 + 1] = cim[t][g];
      }
    }

  // -------- Epilogue: gerelu + trnorm on traces + dense --------------------
  float sr[12], si[12], ssum = 0.f;
#pragma unroll
  for (int u = 0; u < 12; ++u) {
    float r = SCR[lane * 24 + u * 2];
    float q = SCR[lane * 24 + u * 2 + 1];
    float s = fmaxf(r, 0.f);
    float a = s * r, b = s * q;
    sr[u] = a; si[u] = b;
    ssum += sqrtf(a * a + b * b);
  }
  float nrm = fmaxf(ssum * (1.f / 12.f), 0.001f);
  float sc  = 1.f / (3.f * nrm);
  float o = dbg[0];
#pragma unroll
  for (int u = 0; u < 12; ++u)
    o += sc * (sr[u] * dwg[2 * u] + si[u] * dwg[2 * u + 1]);
  if (pt < npts) outg[pt] = o;
}

extern "C" void kernel_launch(void* const* d_in, const int* in_sizes, int n_in,
                              void* d_out, int out_size, void* d_ws, size_t ws_size,
                              hipStream_t stream) {
  const float* x  = (const float*)d_in[0];
  const float* w1 = (const float*)d_in[1];
  const float* w2 = (const float*)d_in[2];
  const float* dw = (const float*)d_in[3];
  const float* db = (const float*)d_in[4];
  float* out = (float*)d_out;
  int npts   = in_sizes[0] / 180;                 // 8192 points
  int blocks = (npts + NPB - 1) / NPB;            // 128 blocks x 64 threads
  size_t shmem = (size_t)(W1F + 2 * PW) * sizeof(float);  // ~99 KB / block
  geblnet_fused<<<blocks, NPB, shmem, stream>>>(x, w1, w2, dw, db, out, npts);
}